// SymMusicMotifGNN_49022756716903
// MI455X (gfx1250) — compile-verified
//
#include <hip/hip_runtime.h>
#include <math.h>

#define NN 20000   // nodes
#define NE 160000  // edges
#define NBATCH 64
#define NH 4
#define NC 128
#define HC 512     // NH*NC

typedef float v2f __attribute__((ext_vector_type(2)));
typedef float v8f __attribute__((ext_vector_type(8)));

__device__ __forceinline__ unsigned fordu(float f) {
  unsigned u = __float_as_uint(f);
  return (u & 0x80000000u) ? ~u : (u | 0x80000000u);
}
__device__ __forceinline__ float uford(unsigned o) {
  return (o & 0x80000000u) ? __uint_as_float(o & 0x7fffffffu)
                           : __uint_as_float(~o);
}
#define ORD_NEG_INF 0x007FFFFFu  // fordu(-inf)

__global__ void fill_u32(unsigned* __restrict__ p, unsigned v, int n) {
  int i = blockIdx.x * blockDim.x + threadIdx.x;
  if (i < n) p[i] = v;
}

// ---------------------------------------------------------------------------
// Node encoder: t = relu(time@ce_w1+b1)@ce_w2+b2 ; h100 = concat(embs, t)
// h0 = relu(LN(h100@ft_w + ft_b))
// one block (128 threads) per node
// ---------------------------------------------------------------------------
__global__ __launch_bounds__(128) void node_encode(
    const float* __restrict__ x,
    const float* __restrict__ emb_inst, const float* __restrict__ emb_pitch,
    const float* __restrict__ emb_index, const float* __restrict__ emb_oct,
    const float* __restrict__ ce_w1, const float* __restrict__ ce_b1,
    const float* __restrict__ ce_w2, const float* __restrict__ ce_b2,
    const float* __restrict__ ft_w, const float* __restrict__ ft_b,
    const float* __restrict__ ft_g, const float* __restrict__ ft_beta,
    float* __restrict__ h0) {
  const int n = blockIdx.x;
  const int t = threadIdx.x;
  __shared__ float sh[128];
  __shared__ float h100[100];
  __shared__ float red[128];
  const float* xrow = x + (size_t)n * 8;
  const int inst  = (int)xrow[0];
  const int pitch = (int)xrow[1];
  const float t0 = xrow[2], t1 = xrow[3], t2 = xrow[4], t3 = xrow[5];
  const int index = (int)xrow[6];
  const int oct   = (int)xrow[7];
  // stage A: hidden = relu(time @ ce_w1 + b1)  (128 wide)
  float hv = ce_b1[t] + t0 * ce_w1[t] + t1 * ce_w1[128 + t] +
             t2 * ce_w1[256 + t] + t3 * ce_w1[384 + t];
  sh[t] = fmaxf(hv, 0.f);
  __syncthreads();
  // stage B: h100[36..99] = hidden @ ce_w2 + b2 ; h100[0..35] = embeddings
  if (t < 64) {
    float acc = ce_b2[t];
#pragma unroll 8
    for (int k = 0; k < 128; ++k) acc += sh[k] * ce_w2[k * 64 + t];
    h100[36 + t] = acc;
  }
  if (t < 8)       h100[t] = emb_inst[inst * 8 + t];
  else if (t < 24) h100[t] = emb_pitch[pitch * 16 + (t - 8)];
  else if (t < 32) h100[t] = emb_index[index * 8 + (t - 24)];
  else if (t < 36) h100[t] = emb_oct[oct * 4 + (t - 32)];
  __syncthreads();
  // stage C: y = h100 @ ft_w + ft_b (128 wide), LN, relu
  float y = ft_b[t];
#pragma unroll 4
  for (int k = 0; k < 100; ++k) y += h100[k] * ft_w[k * 128 + t];
  red[t] = y; __syncthreads();
  for (int s = 64; s > 0; s >>= 1) { if (t < s) red[t] += red[t + s]; __syncthreads(); }
  float mu = red[0] * (1.f / 128.f);
  __syncthreads();
  red[t] = y * y; __syncthreads();
  for (int s = 64; s > 0; s >>= 1) { if (t < s) red[t] += red[t + s]; __syncthreads(); }
  float var = red[0] * (1.f / 128.f) - mu * mu;
  float rs = rsqrtf(var + 1e-5f);
  h0[(size_t)n * 128 + t] = fmaxf(0.f, (y - mu) * rs * ft_g[t] + ft_beta[t]);
}

// ---------------------------------------------------------------------------
// Edge encoder: e32 = relu(edge_attr@ee_w1+b1)@ee_w2+b2   (E x 32)
// ---------------------------------------------------------------------------
__global__ void edge_encode(const float* __restrict__ ea,
                            const float* __restrict__ w1, const float* __restrict__ b1,
                            const float* __restrict__ w2, const float* __restrict__ b2,
                            float* __restrict__ e32) {
  int idx = blockIdx.x * blockDim.x + threadIdx.x;
  if (idx >= NE * 32) return;
  int e = idx >> 5, j = idx & 31;
  float a = ea[e];
  float acc = b2[j];
#pragma unroll
  for (int k = 0; k < 32; ++k)
    acc += fmaxf(a * w1[k] + b1[k], 0.f) * w2[k * 32 + j];
  e32[idx] = acc;
}

// ---------------------------------------------------------------------------
// WMMA f32 GEMM:  C(N x M) = A(N x K) @ W(K x M)
// block = 128 (4 waves); each wave computes a 16x64 output tile using
// V_WMMA_F32_16X16X4_F32, reusing the A fragment across 4 column tiles.
// grid = (N/16, M/256). Requires 16 | N, 256 | M, 4 | K. No divergence
// (EXEC all ones, as WMMA requires).
// ---------------------------------------------------------------------------
__global__ __launch_bounds__(128) void gemm_wmma_f32(
    const float* __restrict__ A, const float* __restrict__ W,
    float* __restrict__ C, int K, int M) {
  const int lane = threadIdx.x & 31;
  const int wave = threadIdx.x >> 5;
  const int half = lane >> 4;   // 0: lanes 0-15, 1: lanes 16-31
  const int l16  = lane & 15;
  const int row0 = blockIdx.x * 16;
  const int col0 = (blockIdx.y * 4 + wave) * 64;
  v8f acc0 = {}, acc1 = {}, acc2 = {}, acc3 = {};
  const float* arow = A + (size_t)(row0 + l16) * K;
  for (int kb = 0; kb < K; kb += 4) {
    // A 16x4 fragment: VGPR0 = K=(kb+2*half), VGPR1 = K=(kb+2*half+1), M = l16
    v2f a;
    const float* ap = arow + kb + 2 * half;
    a.x = ap[0]; a.y = ap[1];
    // B 4x16 fragments (mirror of A layout): N = col + l16
    const float* wp = W + (size_t)(kb + 2 * half) * M + col0 + l16;
    v2f b0; b0.x = wp[0];  b0.y = wp[M + 0];
    v2f b1; b1.x = wp[16]; b1.y = wp[M + 16];
    v2f b2; b2.x = wp[32]; b2.y = wp[M + 32];
    v2f b3; b3.x = wp[48]; b3.y = wp[M + 48];
    acc0 = __builtin_amdgcn_wmma_f32_16x16x4_f32(false, a, false, b0, (short)0, acc0, false, false);
    acc1 = __builtin_amdgcn_wmma_f32_16x16x4_f32(false, a, false, b1, (short)0, acc1, false, false);
    acc2 = __builtin_amdgcn_wmma_f32_16x16x4_f32(false, a, false, b2, (short)0, acc2, false, false);
    acc3 = __builtin_amdgcn_wmma_f32_16x16x4_f32(false, a, false, b3, (short)0, acc3, false, false);
  }
  // C/D layout: VGPR r -> row r (lanes 0-15) / row r+8 (lanes 16-31), col = l16
  float* cp = C + (size_t)(row0 + 8 * half) * M + col0 + l16;
#pragma unroll
  for (int r = 0; r < 8; ++r) {
    cp[(size_t)r * M + 0]  = acc0[r];
    cp[(size_t)r * M + 16] = acc1[r];
    cp[(size_t)r * M + 32] = acc2[r];
    cp[(size_t)r * M + 48] = acc3[r];
  }
}

// ---------------------------------------------------------------------------
// GATv2 pass 1: per (edge, head) logit + segment max (ordered-uint atomicMax)
// ee = e32 @ we computed on the fly.
// ---------------------------------------------------------------------------
__global__ __launch_bounds__(256) void edge_logits(
    const float* __restrict__ xl, const float* __restrict__ xr,
    const float* __restrict__ e32,
    const float* __restrict__ we, const float* __restrict__ att,
    const int* __restrict__ src, const int* __restrict__ dst,
    float* __restrict__ logits, unsigned* __restrict__ mOrd) {
  int idx = blockIdx.x * blockDim.x + threadIdx.x;
  if (idx >= NE * NH) return;
  int e = idx >> 2, h = idx & 3;
  int s = src[e], d = dst[e];
  const float* xlp = xl + (size_t)s * HC + h * NC;
  const float* xrp = xr + (size_t)d * HC + h * NC;
  const float* ep = e32 + (size_t)e * 32;
  float er[32];
#pragma unroll
  for (int k = 0; k < 32; ++k) er[k] = ep[k];
  const float* wp = we + h * NC;
  const float* ap = att + h * NC;
  float logit = 0.f;
  for (int c = 0; c < NC; ++c) {
    float ee = 0.f;
#pragma unroll
    for (int k = 0; k < 32; ++k) ee += er[k] * wp[k * HC + c];
    float g = xlp[c] + xrp[c] + ee;
    g = (g > 0.f) ? g : 0.2f * g;
    logit += g * ap[c];
  }
  logits[idx] = logit;
  atomicMax(mOrd + (size_t)d * NH + h, fordu(logit));
}

// ---------------------------------------------------------------------------
// GATv2 pass 2: a = exp(logit - m); den += a; num[dst] += a * xl[src]
// ---------------------------------------------------------------------------
__global__ __launch_bounds__(256) void edge_accum(
    const float* __restrict__ xl,
    const int* __restrict__ src, const int* __restrict__ dst,
    const float* __restrict__ logits, const unsigned* __restrict__ mOrd,
    float* __restrict__ den, float* __restrict__ num) {
  int idx = blockIdx.x * blockDim.x + threadIdx.x;
  if (idx >= NE * NH) return;
  int e = idx >> 2, h = idx & 3;
  int s = src[e], d = dst[e];
  float m = uford(mOrd[(size_t)d * NH + h]);
  float a = expf(logits[idx] - m);
  atomicAdd(den + (size_t)d * NH + h, a);
  const float* xlp = xl + (size_t)s * HC + h * NC;
  float* np = num + (size_t)d * HC + h * NC;
#pragma unroll 4
  for (int c = 0; c < NC; ++c) atomicAdd(np + c, a * xlp[c]);
}

// ---------------------------------------------------------------------------
// GATv2 layer1 finalize: h1 = relu(LN(num/den + bias))   (512 wide, concat)
// block = 256 threads, 2 channels each
// ---------------------------------------------------------------------------
__global__ __launch_bounds__(256) void node_ln1(
    const float* __restrict__ num, const float* __restrict__ den,
    const float* __restrict__ bias, const float* __restrict__ g,
    const float* __restrict__ beta, float* __restrict__ h1) {
  int n = blockIdx.x, t = threadIdx.x;
  __shared__ float red[256];
  float d0 = den[n * 4 + (t >> 7)] + 1e-16f;
  float d1 = den[n * 4 + ((t + 256) >> 7)] + 1e-16f;
  float v0 = num[(size_t)n * HC + t]       / d0 + bias[t];
  float v1 = num[(size_t)n * HC + 256 + t] / d1 + bias[256 + t];
  red[t] = v0 + v1; __syncthreads();
  for (int s = 128; s > 0; s >>= 1) { if (t < s) red[t] += red[t + s]; __syncthreads(); }
  float mu = red[0] * (1.f / 512.f); __syncthreads();
  red[t] = v0 * v0 + v1 * v1; __syncthreads();
  for (int s = 128; s > 0; s >>= 1) { if (t < s) red[t] += red[t + s]; __syncthreads(); }
  float var = red[0] * (1.f / 512.f) - mu * mu;
  float rs = rsqrtf(var + 1e-5f);
  h1[(size_t)n * HC + t]       = fmaxf(0.f, (v0 - mu) * rs * g[t] + beta[t]);
  h1[(size_t)n * HC + 256 + t] = fmaxf(0.f, (v1 - mu) * rs * g[256 + t] + beta[256 + t]);
}

// ---------------------------------------------------------------------------
// GATv2 layer2 finalize: mean over heads, +bias, LN, +identity(h0), relu
// ---------------------------------------------------------------------------
__global__ __launch_bounds__(128) void node_ln2(
    const float* __restrict__ num, const float* __restrict__ den,
    const float* __restrict__ bias, const float* __restrict__ g,
    const float* __restrict__ beta, const float* __restrict__ h0,
    float* __restrict__ h2) {
  int n = blockIdx.x, t = threadIdx.x;
  __shared__ float red[128];
  float v = 0.f;
#pragma unroll
  for (int h = 0; h < NH; ++h)
    v += num[(size_t)n * HC + h * NC + t] / (den[n * 4 + h] + 1e-16f);
  v = 0.25f * v + bias[t];
  red[t] = v; __syncthreads();
  for (int s = 64; s > 0; s >>= 1) { if (t < s) red[t] += red[t + s]; __syncthreads(); }
  float mu = red[0] * (1.f / 128.f); __syncthreads();
  red[t] = v * v; __syncthreads();
  for (int s = 64; s > 0; s >>= 1) { if (t < s) red[t] += red[t + s]; __syncthreads(); }
  float var = red[0] * (1.f / 128.f) - mu * mu;
  float rs = rsqrtf(var + 1e-5f);
  float ln = (v - mu) * rs * g[t] + beta[t];
  h2[(size_t)n * 128 + t] = fmaxf(0.f, ln + h0[(size_t)n * 128 + t]);
}

// ---------------------------------------------------------------------------
// gate = tanh(h2@ro_w1+b1)@ro_w2+b2  (scalar per node); block = 64
// ---------------------------------------------------------------------------
__global__ __launch_bounds__(64) void gate_kernel(
    const float* __restrict__ h2, const float* __restrict__ w1,
    const float* __restrict__ b1, const float* __restrict__ w2,
    const float* __restrict__ b2, float* __restrict__ gate) {
  int n = blockIdx.x, j = threadIdx.x;
  __shared__ float red[64];
  float acc = b1[j];
  const float* hp = h2 + (size_t)n * 128;
#pragma unroll 4
  for (int c = 0; c < 128; ++c) acc += hp[c] * w1[c * 64 + j];
  red[j] = tanhf(acc) * w2[j];
  __syncthreads();
  for (int s = 32; s > 0; s >>= 1) { if (j < s) red[j] += red[j + s]; __syncthreads(); }
  if (j == 0) gate[n] = red[0] + b2[0];
}

__global__ void poolA(const int* __restrict__ batch, const float* __restrict__ gate,
                      unsigned* __restrict__ gmOrd, float* __restrict__ counts) {
  int i = blockIdx.x * blockDim.x + threadIdx.x;
  if (i >= NN) return;
  int b = batch[i];
  atomicMax(gmOrd + b, fordu(gate[i]));
  atomicAdd(counts + b, 1.f);
}

__global__ void poolC(const int* __restrict__ batch, float* __restrict__ gate,
                      const unsigned* __restrict__ gmOrd, float* __restrict__ gden) {
  int i = blockIdx.x * blockDim.x + threadIdx.x;
  if (i >= NN) return;
  int b = batch[i];
  float ge = expf(gate[i] - uford(gmOrd[b]));
  gate[i] = ge;  // overwrite with exp value
  atomicAdd(gden + b, ge);
}

__global__ void poolD(const int* __restrict__ batch, const float* __restrict__ h2,
                      const float* __restrict__ ge, const float* __restrict__ gden,
                      float* __restrict__ attacc, float* __restrict__ meanacc,
                      unsigned* __restrict__ maxOrd) {
  int idx = blockIdx.x * blockDim.x + threadIdx.x;
  if (idx >= NN * NC) return;
  int n = idx >> 7, c = idx & 127;
  int b = batch[n];
  float hv = h2[(size_t)n * 128 + c];
  float w = ge[n] / (gden[b] + 1e-16f);
  atomicAdd(attacc + (size_t)b * 128 + c, w * hv);
  atomicAdd(meanacc + (size_t)b * 128 + c, hv);
  atomicMax(maxOrd + (size_t)b * 128 + c, fordu(hv));
}

// ---------------------------------------------------------------------------
// final head: z = [x_att, x_mean, x_max] (384); z2 = relu(LN(z@fp_w1+b1));
// out = softplus(z2@fp_w2+b2)   (B x 64)
// ---------------------------------------------------------------------------
__global__ __launch_bounds__(128) void final_kernel(
    const float* __restrict__ attacc, const float* __restrict__ meanacc,
    const unsigned* __restrict__ maxOrd, const float* __restrict__ counts,
    const float* __restrict__ fp_w1, const float* __restrict__ fp_b1,
    const float* __restrict__ fp_g, const float* __restrict__ fp_beta,
    const float* __restrict__ fp_w2, const float* __restrict__ fp_b2,
    float* __restrict__ out) {
  int b = blockIdx.x, t = threadIdx.x;
  __shared__ float z[384];
  __shared__ float z2[128];
  __shared__ float red[128];
  float cnt = counts[b];
  z[t] = attacc[(size_t)b * 128 + t];
  z[128 + t] = meanacc[(size_t)b * 128 + t] / fmaxf(cnt, 1.f);
  float mx = uford(maxOrd[(size_t)b * 128 + t]);
  z[256 + t] = (cnt > 0.f) ? mx : 0.f;
  __syncthreads();
  float v = fp_b1[t];
#pragma unroll 4
  for (int k = 0; k < 384; ++k) v += z[k] * fp_w1[k * 128 + t];
  red[t] = v; __syncthreads();
  for (int s = 64; s > 0; s >>= 1) { if (t < s) red[t] += red[t + s]; __syncthreads(); }
  float mu = red[0] * (1.f / 128.f); __syncthreads();
  red[t] = v * v; __syncthreads();
  for (int s = 64; s > 0; s >>= 1) { if (t < s) red[t] += red[t + s]; __syncthreads(); }
  float var = red[0] * (1.f / 128.f) - mu * mu;
  float rs = rsqrtf(var + 1e-5f);
  z2[t] = fmaxf(0.f, (v - mu) * rs * fp_g[t] + fp_beta[t]);
  __syncthreads();
  if (t < 64) {
    float a = fp_b2[t];
#pragma unroll 4
    for (int k = 0; k < 128; ++k) a += z2[k] * fp_w2[k * 64 + t];
    out[(size_t)b * 64 + t] = fmaxf(a, 0.f) + log1pf(expf(-fabsf(a)));
  }
}

// ---------------------------------------------------------------------------
extern "C" void kernel_launch(void* const* d_in, const int* in_sizes, int n_in,
                              void* d_out, int out_size, void* d_ws, size_t ws_size,
                              hipStream_t stream) {
  const float* x         = (const float*)d_in[0];
  const int*   eidx      = (const int*)  d_in[1];
  const float* edge_attr = (const float*)d_in[2];
  const int*   batch     = (const int*)  d_in[3];
  const float* emb_inst  = (const float*)d_in[4];
  const float* emb_pitch = (const float*)d_in[5];
  const float* emb_index = (const float*)d_in[6];
  const float* emb_oct   = (const float*)d_in[7];
  const float* ce_w1     = (const float*)d_in[8];
  const float* ce_b1     = (const float*)d_in[9];
  const float* ce_w2     = (const float*)d_in[10];
  const float* ce_b2     = (const float*)d_in[11];
  const float* ft_w      = (const float*)d_in[12];
  const float* ft_b      = (const float*)d_in[13];
  const float* ft_g      = (const float*)d_in[14];
  const float* ft_beta   = (const float*)d_in[15];
  const float* ee_w1     = (const float*)d_in[16];
  const float* ee_b1     = (const float*)d_in[17];
  const float* ee_w2     = (const float*)d_in[18];
  const float* ee_b2     = (const float*)d_in[19];
  const float* c1_wl     = (const float*)d_in[20];
  const float* c1_wr     = (const float*)d_in[21];
  const float* c1_we     = (const float*)d_in[22];
  const float* c1_att    = (const float*)d_in[23];
  const float* c1_b      = (const float*)d_in[24];
  const float* n1_g      = (const float*)d_in[25];
  const float* n1_beta   = (const float*)d_in[26];
  const float* c2_wl     = (const float*)d_in[27];
  const float* c2_wr     = (const float*)d_in[28];
  const float* c2_we     = (const float*)d_in[29];
  const float* c2_att    = (const float*)d_in[30];
  const float* c2_b      = (const float*)d_in[31];
  const float* n2_g      = (const float*)d_in[32];
  const float* n2_beta   = (const float*)d_in[33];
  const float* ro_w1     = (const float*)d_in[34];
  const float* ro_b1     = (const float*)d_in[35];
  const float* ro_w2     = (const float*)d_in[36];
  const float* ro_b2     = (const float*)d_in[37];
  const float* fp_w1     = (const float*)d_in[38];
  const float* fp_b1     = (const float*)d_in[39];
  const float* fp_g      = (const float*)d_in[40];
  const float* fp_beta   = (const float*)d_in[41];
  const float* fp_w2     = (const float*)d_in[42];
  const float* fp_b2     = (const float*)d_in[43];

  const int* src = eidx;
  const int* dst = eidx + NE;

  // -------- workspace carve-out (all 256B aligned) --------
  char* base = (char*)d_ws;
  size_t off = 0;
  auto alloc = [&](size_t nbytes) -> void* {
    void* p = base + off;
    off += (nbytes + 255) & ~(size_t)255;
    return p;
  };
  float*    h0      = (float*)   alloc((size_t)NN * 128 * 4);
  float*    e32     = (float*)   alloc((size_t)NE * 32 * 4);
  float*    xl      = (float*)   alloc((size_t)NN * HC * 4);
  float*    xr      = (float*)   alloc((size_t)NN * HC * 4);
  float*    num     = (float*)   alloc((size_t)NN * HC * 4);
  float*    h1      = (float*)   alloc((size_t)NN * HC * 4);
  float*    h2      = (float*)   alloc((size_t)NN * 128 * 4);
  float*    logits  = (float*)   alloc((size_t)NE * NH * 4);
  unsigned* mOrd    = (unsigned*)alloc((size_t)NN * NH * 4);
  float*    den     = (float*)   alloc((size_t)NN * NH * 4);
  float*    gate    = (float*)   alloc((size_t)NN * 4);
  unsigned* gmOrd   = (unsigned*)alloc((size_t)NBATCH * 4);
  float*    gden    = (float*)   alloc((size_t)NBATCH * 4);
  float*    counts  = (float*)   alloc((size_t)NBATCH * 4);
  float*    attacc  = (float*)   alloc((size_t)NBATCH * 128 * 4);
  float*    meanacc = (float*)   alloc((size_t)NBATCH * 128 * 4);
  unsigned* maxOrd  = (unsigned*)alloc((size_t)NBATCH * 128 * 4);
  (void)ws_size; (void)in_sizes; (void)n_in; (void)out_size;

  auto fill = [&](void* p, unsigned v, int n) {
    fill_u32<<<(n + 255) / 256, 256, 0, stream>>>((unsigned*)p, v, n);
  };

  // -------- encoders --------
  node_encode<<<NN, 128, 0, stream>>>(x, emb_inst, emb_pitch, emb_index, emb_oct,
                                      ce_w1, ce_b1, ce_w2, ce_b2,
                                      ft_w, ft_b, ft_g, ft_beta, h0);
  edge_encode<<<(NE * 32 + 255) / 256, 256, 0, stream>>>(edge_attr, ee_w1, ee_b1,
                                                         ee_w2, ee_b2, e32);

  const dim3 gemm_grid(NN / 16, HC / 256);  // 1250 x 2

  // -------- GAT layer 1 (concat) --------
  fill(num, 0u, NN * HC);
  fill(mOrd, ORD_NEG_INF, NN * NH);
  fill(den, 0u, NN * NH);
  gemm_wmma_f32<<<gemm_grid, 128, 0, stream>>>(h0, c1_wl, xl, 128, HC);
  gemm_wmma_f32<<<gemm_grid, 128, 0, stream>>>(h0, c1_wr, xr, 128, HC);
  edge_logits<<<(NE * NH + 255) / 256, 256, 0, stream>>>(xl, xr, e32, c1_we, c1_att,
                                                         src, dst, logits, mOrd);
  edge_accum<<<(NE * NH + 255) / 256, 256, 0, stream>>>(xl, src, dst, logits, mOrd,
                                                        den, num);
  node_ln1<<<NN, 256, 0, stream>>>(num, den, c1_b, n1_g, n1_beta, h1);

  // -------- GAT layer 2 (mean over heads) --------
  fill(num, 0u, NN * HC);
  fill(mOrd, ORD_NEG_INF, NN * NH);
  fill(den, 0u, NN * NH);
  gemm_wmma_f32<<<gemm_grid, 128, 0, stream>>>(h1, c2_wl, xl, HC, HC);
  gemm_wmma_f32<<<gemm_grid, 128, 0, stream>>>(h1, c2_wr, xr, HC, HC);
  edge_logits<<<(NE * NH + 255) / 256, 256, 0, stream>>>(xl, xr, e32, c2_we, c2_att,
                                                         src, dst, logits, mOrd);
  edge_accum<<<(NE * NH + 255) / 256, 256, 0, stream>>>(xl, src, dst, logits, mOrd,
                                                        den, num);
  node_ln2<<<NN, 128, 0, stream>>>(num, den, c2_b, n2_g, n2_beta, h0, h2);

  // -------- readout + pooling --------
  gate_kernel<<<NN, 64, 0, stream>>>(h2, ro_w1, ro_b1, ro_w2, ro_b2, gate);
  fill(gmOrd, ORD_NEG_INF, NBATCH);
  fill(gden, 0u, NBATCH);
  fill(counts, 0u, NBATCH);
  fill(attacc, 0u, NBATCH * 128);
  fill(meanacc, 0u, NBATCH * 128);
  fill(maxOrd, ORD_NEG_INF, NBATCH * 128);
  poolA<<<(NN + 255) / 256, 256, 0, stream>>>(batch, gate, gmOrd, counts);
  poolC<<<(NN + 255) / 256, 256, 0, stream>>>(batch, gate, gmOrd, gden);
  poolD<<<(NN * NC + 255) / 256, 256, 0, stream>>>(batch, h2, gate, gden,
                                                   attacc, meanacc, maxOrd);
  final_kernel<<<NBATCH, 128, 0, stream>>>(attacc, meanacc, maxOrd, counts,
                                           fp_w1, fp_b1, fp_g, fp_beta,
                                           fp_w2, fp_b2, (float*)d_out);
}